// Model_82721070121711
// MI455X (gfx1250) — compile-verified
//
#include <hip/hip_runtime.h>

// ---------------------------------------------------------------------------
// f32 WMMA helpers (CDNA5 gfx1250, wave32): D(16x16,f32) = A(16x4,f32) x B(4x16,f32) + C
// Per-lane fragments (ISA 7.12.2):
//   A: v2f, element v -> (row = lane&15, k = (lane>>4)*2 + v)   [k contiguous]
//   B: v2f, element v -> (col = lane&15, k = (lane>>4)*2 + v)   [k contiguous, col-major arr[n][k]]
//   C/D: v8f, element r -> (row = r + (lane>>4)*8, col = lane&15)
// ---------------------------------------------------------------------------
typedef float v2f __attribute__((ext_vector_type(2)));
typedef float v8f __attribute__((ext_vector_type(8)));

__device__ __forceinline__ v8f wmma4(v2f a, v2f b, v8f c) {
  return __builtin_amdgcn_wmma_f32_16x16x4_f32(false, a, false, b, (short)0, c,
                                               false, false);
}

__device__ __forceinline__ v8f zf8() {
  v8f z = {0.f, 0.f, 0.f, 0.f, 0.f, 0.f, 0.f, 0.f};
  return z;
}

// ---------------------------------------------------------------------------
// Model constants
// ---------------------------------------------------------------------------
#define NB   128   // graphs
#define NPG  32    // nodes per graph
#define NN   4096  // total nodes
#define DIM  64
#define RW   6
#define KXC  80    // padded D+RW (70 -> 80)
#define H1D  128
#define H2D  64
#define FCH  40    // PPGN channels
#define FPAD 48    // padded channels (3 tiles of 16)
#define NPIX 1024  // 32*32
#define KHOP_SPLIT 8

// ===========================================================================
// prep kernels
// ===========================================================================
__global__ void k_zero(float* p, int n) {
  int i = blockIdx.x * blockDim.x + threadIdx.x;
  if (i < n) p[i] = 0.f;
}

// Xc [4096][80] (padded) and XcT [80][4096]
__global__ void k_prep_xc(const float* x, const float* wf, float* Xc, float* XcT) {
  int i = blockIdx.x * blockDim.x + threadIdx.x;
  if (i >= NN * KXC) return;
  int m = i / KXC, c = i % KXC;
  float v = 0.f;
  if (c < DIM)            v = x[m * DIM + c];
  else if (c < DIM + RW)  v = wf[m * RW + (c - DIM)];
  Xc[i] = v;
  XcT[(size_t)c * NN + m] = v;
}

// M[b][pix] = Mmask[b][0]+Mmask[b][1]; h0 channel-last [b][pix][4] from X2 [b][3][pix]
__global__ void k_prep_mh0(const float* Mmask, const float* X2, float* Mv, float* h0) {
  int i = blockIdx.x * blockDim.x + threadIdx.x;
  if (i >= NB * NPIX) return;
  int b = i >> 10, pix = i & (NPIX - 1);
  Mv[i] = Mmask[((size_t)b * 2) * NPIX + pix] + Mmask[((size_t)b * 2 + 1) * NPIX + pix];
#pragma unroll
  for (int ci = 0; ci < 4; ci++)
    h0[(size_t)i * 4 + ci] = (ci < 3) ? X2[((size_t)b * 3 + ci) * NPIX + pix] : 0.f;
}

// dst[48][ldK], dst[n][k] = (n<srcN && k<Kvalid) ? src[n*srcCi + colOff + k] : 0
__global__ void k_pack_w(float* dst, const float* src, int srcN, int srcCi,
                         int colOff, int ldK, int Kvalid) {
  int i = blockIdx.x * blockDim.x + threadIdx.x;
  if (i >= FPAD * ldK) return;
  int n = i / ldK, k = i % ldK;
  dst[i] = (n < srcN && k < Kvalid) ? src[n * srcCi + colOff + k] : 0.f;
}

// dst[N][ldK], dst[n][k] = (k<Ksrc) ? src[k*N + n] : 0   (transpose+pad)
__global__ void k_pack_T(float* dst, const float* src, int Ksrc, int N, int ldK) {
  int i = blockIdx.x * blockDim.x + threadIdx.x;
  if (i >= N * ldK) return;
  int n = i / ldK, k = i % ldK;
  dst[i] = (k < Ksrc) ? src[(size_t)k * N + n] : 0.f;
}

// ===========================================================================
// Y_z = hop_z @ Xc   (M=4096, N=80, K=4096), K split 8 ways, atomic accumulate
// grid (64, 8, 3), block 128 (4 waves); wave mtile = bx*4+wv
// ===========================================================================
__global__ void k_hop_gemm(const float* hop1, const float* hop2, const float* hop3,
                           const float* XcT, float* Y) {
  int wv = threadIdx.x >> 5, lane = threadIdx.x & 31;
  int mtile = blockIdx.x * 4 + wv;
  const float* A = (blockIdx.z == 0) ? hop1 : (blockIdx.z == 1) ? hop2 : hop3;
  float* Yp = Y + (size_t)blockIdx.z * NN * KXC;
  int m0 = mtile * 16;
  int r = lane & 15, ko = (lane >> 4) * 2;
  const int KC = NN / KHOP_SPLIT;
  int kbeg = blockIdx.y * KC;
  const float* Arow = A + (size_t)(m0 + r) * NN + ko;
  v8f acc[5];
#pragma unroll
  for (int nt = 0; nt < 5; nt++) acc[nt] = zf8();
  for (int k = kbeg; k < kbeg + KC; k += 4) {
    v2f a = *(const v2f*)(Arow + k);
#pragma unroll
    for (int nt = 0; nt < 5; nt++) {
      v2f b = *(const v2f*)(XcT + (size_t)(nt * 16 + r) * NN + k + ko);
      acc[nt] = wmma4(a, b, acc[nt]);
    }
  }
  int cl = lane & 15, rb = (lane >> 4) * 8;
#pragma unroll
  for (int nt = 0; nt < 5; nt++)
#pragma unroll
    for (int rr = 0; rr < 8; rr++)
      atomicAdd(&Yp[(size_t)(m0 + rb + rr) * KXC + nt * 16 + cl], acc[nt][rr]);
}

// ===========================================================================
// Row GEMM: Out[m][n] = act(sum_k A[m][k]*W[n][k] + bias[n]) (+= if ACCUM)
// A row-major (lda), W packed [NT*16][K] k-contiguous. grid.x*4 waves cover rows/16.
// ===========================================================================
template <int NT, int RELU, int ACCUM>
__global__ void k_gemm_rows(const float* __restrict__ A, int lda,
                            const float* __restrict__ W, int K,
                            const float* __restrict__ bias,
                            float* Out, int ldo) {
  int wv = threadIdx.x >> 5, lane = threadIdx.x & 31;
  int m0 = (blockIdx.x * 4 + wv) * 16;
  int r = lane & 15, ko = (lane >> 4) * 2;
  v8f acc[NT];
#pragma unroll
  for (int nt = 0; nt < NT; nt++) acc[nt] = zf8();
  const float* Ar = A + (size_t)(m0 + r) * lda + ko;
  for (int k = 0; k < K; k += 4) {
    v2f a = *(const v2f*)(Ar + k);
#pragma unroll
    for (int nt = 0; nt < NT; nt++) {
      v2f b = *(const v2f*)(W + (size_t)(nt * 16 + r) * K + k + ko);
      acc[nt] = wmma4(a, b, acc[nt]);
    }
  }
  int cl = lane & 15, rb = (lane >> 4) * 8;
#pragma unroll
  for (int nt = 0; nt < NT; nt++) {
    float bv = bias[nt * 16 + cl];
#pragma unroll
    for (int rr = 0; rr < 8; rr++) {
      float v = acc[nt][rr] + bv;
      if (RELU) v = fmaxf(v, 0.f);
      float* o = &Out[(size_t)(m0 + rb + rr) * ldo + nt * 16 + cl];
      if (ACCUM) v += *o;
      *o = v;
    }
  }
}

// ===========================================================================
// PPGN 1x1 conv: out = relu(M * (A1 @ W1^T [+ A2 @ W2^T]))
// A channel-last [b][1024][lda]; W [48][K] k-contiguous.
// mode 0: out[b][pix][48] ; mode 1: out[b][co][i][k]=v(pix=i*32+k) ;
// mode 2: out[b][co][j][k] (transposed pixel matrix)
// grid (48, 128), block 128; wave w = bx*4+wv : ptile=w/3 (64), cotile=w%3
// ===========================================================================
__global__ void k_conv(const float* A1, int lda1, int K1, const float* W1,
                       const float* A2, int lda2, int K2, const float* W2,
                       const float* Mv, float* Out, int mode) {
  int wv = threadIdx.x >> 5, lane = threadIdx.x & 31;
  int w = blockIdx.x * 4 + wv;
  int pt = w / 3, ct = w % 3;
  int b = blockIdx.y;
  int m0 = pt * 16;
  int r = lane & 15, ko = (lane >> 4) * 2;
  v8f acc = zf8();
  {
    const float* Ar = A1 + ((size_t)b * NPIX + m0 + r) * lda1 + ko;
    for (int k = 0; k < K1; k += 4) {
      v2f a = *(const v2f*)(Ar + k);
      v2f bb = *(const v2f*)(W1 + (size_t)(ct * 16 + r) * K1 + k + ko);
      acc = wmma4(a, bb, acc);
    }
  }
  if (A2) {
    const float* Ar = A2 + ((size_t)b * NPIX + m0 + r) * lda2 + ko;
    for (int k = 0; k < K2; k += 4) {
      v2f a = *(const v2f*)(Ar + k);
      v2f bb = *(const v2f*)(W2 + (size_t)(ct * 16 + r) * K2 + k + ko);
      acc = wmma4(a, bb, acc);
    }
  }
  int cl = lane & 15, rb = (lane >> 4) * 8;
  int co = ct * 16 + cl;
#pragma unroll
  for (int rr = 0; rr < 8; rr++) {
    int pix = m0 + rb + rr;
    float v = fmaxf(acc[rr] * Mv[(size_t)b * NPIX + pix], 0.f);
    size_t idx;
    if (mode == 0)      idx = ((size_t)b * NPIX + pix) * FPAD + co;
    else if (mode == 1) idx = ((size_t)b * FPAD + co) * NPIX + pix;
    else                idx = ((size_t)b * FPAD + co) * NPIX + ((pix & 31) * 32 + (pix >> 5));
    Out[idx] = v;
  }
}

// ===========================================================================
// Batched bmm: x12[b][pix][c] = M[b][pix] * sum_k x1[b][c][i][k]*x2T[b][c][j][k]
// grid (40, 128), block 128; wave w: c=w>>2, it=(w>>1)&1, jt=w&1
// ===========================================================================
__global__ void k_bmm(const float* x1, const float* x2t, const float* Mv, float* x12) {
  int wv = threadIdx.x >> 5, lane = threadIdx.x & 31;
  int w = blockIdx.x * 4 + wv;
  int c = w >> 2, it = (w >> 1) & 1, jt = w & 1;
  int b = blockIdx.y;
  const float* X1 = x1 + ((size_t)b * FPAD + c) * NPIX;
  const float* X2 = x2t + ((size_t)b * FPAD + c) * NPIX;
  int r = lane & 15, ko = (lane >> 4) * 2;
  v8f acc = zf8();
#pragma unroll
  for (int k = 0; k < 32; k += 4) {
    v2f a  = *(const v2f*)(X1 + (it * 16 + r) * 32 + k + ko);
    v2f bb = *(const v2f*)(X2 + (jt * 16 + r) * 32 + k + ko);
    acc = wmma4(a, bb, acc);
  }
  int cl = lane & 15, rb = (lane >> 4) * 8;
#pragma unroll
  for (int rr = 0; rr < 8; rr++) {
    int i = it * 16 + rb + rr, j = jt * 16 + cl, pix = i * 32 + j;
    x12[((size_t)b * NPIX + pix) * FPAD + c] = acc[rr] * Mv[(size_t)b * NPIX + pix];
  }
}

// readout: outs[b][l*80 + which*40 + co] = sum_pix h[b][pix][co]*Mmask[b][which][pix]
__global__ void k_readout(const float* h, const float* Mmask, float* outs, int l) {
  int b = blockIdx.x, t = threadIdx.x;
  if (t >= 2 * FCH) return;
  int which = t / FCH, co = t % FCH;
  const float* hp = h + (size_t)b * NPIX * FPAD + co;
  const float* mp = Mmask + ((size_t)b * 2 + which) * NPIX;
  float s = 0.f;
  for (int pix = 0; pix < NPIX; pix++) s += hp[(size_t)pix * FPAD] * mp[pix];
  outs[(size_t)b * 400 + l * 80 + which * FCH + co] = s;
}

// segment sum over 32 consecutive nodes -> out cols 32..95
__global__ void k_segsum(const float* sub, float* out) {
  int b = blockIdx.x, t = threadIdx.x;  // 64 threads
  float s = 0.f;
  for (int rr = 0; rr < NPG; rr++) s += sub[((size_t)(b * NPG + rr)) * H2D + t];
  out[(size_t)b * 96 + 32 + t] = s;
}

// ===========================================================================
extern "C" void kernel_launch(void* const* d_in, const int* in_sizes, int n_in,
                              void* d_out, int out_size, void* d_ws, size_t ws_size,
                              hipStream_t stream) {
  (void)in_sizes; (void)n_in; (void)out_size; (void)ws_size;
  const float* x     = (const float*)d_in[0];
  const float* wf    = (const float*)d_in[1];
  const float* hop1  = (const float*)d_in[2];
  const float* hop2  = (const float*)d_in[3];
  const float* hop3  = (const float*)d_in[4];
  const float* X2    = (const float*)d_in[5];
  const float* Mmask = (const float*)d_in[6];
  const float* pa[5], *pb[5], *pc[5];
  for (int l = 0; l < 5; l++) {
    pa[l] = (const float*)d_in[7 + l * 3 + 0];
    pb[l] = (const float*)d_in[7 + l * 3 + 1];
    pc[l] = (const float*)d_in[7 + l * 3 + 2];
  }
  const float* h1w = (const float*)d_in[22];
  const float* h1b = (const float*)d_in[23];
  const float* sw1[4], *sb1[4], *sw2[4], *sb2[4];
  for (int l = 0; l < 4; l++) {
    sw1[l] = (const float*)d_in[24 + l * 4 + 0];
    sb1[l] = (const float*)d_in[24 + l * 4 + 1];
    sw2[l] = (const float*)d_in[24 + l * 4 + 2];
    sb2[l] = (const float*)d_in[24 + l * 4 + 3];
  }
  float* out = (float*)d_out;

  // ---- workspace layout (floats, 64-elem aligned) ----
  float* W = (float*)d_ws;
  size_t off = 0;
  auto alc = [&](size_t n) { size_t o = off; off += (n + 63) & ~(size_t)63; return o; };
  size_t oXc  = alc((size_t)NN * KXC);
  size_t oXcT = alc((size_t)KXC * NN);
  size_t oY   = alc((size_t)3 * NN * KXC);
  size_t oH   = alc((size_t)NN * H1D);
  size_t oSub = alc((size_t)NN * H2D);
  size_t oM   = alc((size_t)NB * NPIX);
  size_t oH0  = alc((size_t)NB * NPIX * 4);
  size_t oP[4];
  for (int i = 0; i < 4; i++) oP[i] = alc((size_t)NB * NPIX * FPAD);
  size_t oOuts = alc((size_t)NB * 400);
  size_t oH1wT = alc((size_t)32 * 400);
  size_t owa[5], owb[5], owc1[5], owc2[5];
  owa[0] = alc(FPAD * 4);  owb[0] = alc(FPAD * 4);
  owc1[0] = alc(FPAD * 40); owc2[0] = alc(FPAD * 4);
  for (int l = 1; l < 5; l++) {
    owa[l] = alc(FPAD * 40); owb[l] = alc(FPAD * 40);
    owc1[l] = alc(FPAD * 40); owc2[l] = alc(FPAD * 40);
  }
  size_t ow1[4], ow2[4];
  for (int l = 0; l < 4; l++) { ow1[l] = alc((size_t)H1D * KXC); ow2[l] = alc((size_t)H2D * H1D); }

  auto G = [](int n) { return dim3((n + 255) / 256); };

  // ---- prep ----
  k_zero<<<G(3 * NN * KXC), 256, 0, stream>>>(W + oY, 3 * NN * KXC);
  k_prep_xc<<<G(NN * KXC), 256, 0, stream>>>(x, wf, W + oXc, W + oXcT);
  k_prep_mh0<<<G(NB * NPIX), 256, 0, stream>>>(Mmask, X2, W + oM, W + oH0);
  // PPGN weights
  k_pack_w<<<G(FPAD * 4), 256, 0, stream>>>(W + owa[0], pa[0], FCH, 3, 0, 4, 3);
  k_pack_w<<<G(FPAD * 4), 256, 0, stream>>>(W + owb[0], pb[0], FCH, 3, 0, 4, 3);
  k_pack_w<<<G(FPAD * 40), 256, 0, stream>>>(W + owc1[0], pc[0], FCH, 43, 0, 40, 40);
  k_pack_w<<<G(FPAD * 4), 256, 0, stream>>>(W + owc2[0], pc[0], FCH, 43, 40, 4, 3);
  for (int l = 1; l < 5; l++) {
    k_pack_w<<<G(FPAD * 40), 256, 0, stream>>>(W + owa[l], pa[l], FCH, 40, 0, 40, 40);
    k_pack_w<<<G(FPAD * 40), 256, 0, stream>>>(W + owb[l], pb[l], FCH, 40, 0, 40, 40);
    k_pack_w<<<G(FPAD * 40), 256, 0, stream>>>(W + owc1[l], pc[l], FCH, 80, 0, 40, 40);
    k_pack_w<<<G(FPAD * 40), 256, 0, stream>>>(W + owc2[l], pc[l], FCH, 80, 40, 40, 40);
  }
  // MLP weights (transpose to [n][k]) and head weight
  for (int l = 0; l < 4; l++) {
    k_pack_T<<<G(H1D * KXC), 256, 0, stream>>>(W + ow1[l], sw1[l], 70, H1D, KXC);
    k_pack_T<<<G(H2D * H1D), 256, 0, stream>>>(W + ow2[l], sw2[l], H1D, H2D, H1D);
  }
  k_pack_T<<<G(32 * 400), 256, 0, stream>>>(W + oH1wT, h1w, 400, 32, 400);

  // ---- subgraph branch ----
  k_hop_gemm<<<dim3(64, KHOP_SPLIT, 3), 128, 0, stream>>>(hop1, hop2, hop3,
                                                          W + oXcT, W + oY);
  for (int l = 0; l < 4; l++) {
    const float* Ain = (l == 0) ? (W + oXc) : (W + oY + (size_t)(l - 1) * NN * KXC);
    k_gemm_rows<8, 1, 0><<<64, 128, 0, stream>>>(Ain, KXC, W + ow1[l], KXC,
                                                 sb1[l], W + oH, H1D);
    if (l == 0)
      k_gemm_rows<4, 0, 0><<<64, 128, 0, stream>>>(W + oH, H1D, W + ow2[l], H1D,
                                                   sb2[l], W + oSub, H2D);
    else
      k_gemm_rows<4, 0, 1><<<64, 128, 0, stream>>>(W + oH, H1D, W + ow2[l], H1D,
                                                   sb2[l], W + oSub, H2D);
  }
  k_segsum<<<NB, 64, 0, stream>>>(W + oSub, out);

  // ---- PPGN branch ----
  int ih = -1;  // -1 => h0 (lda 4, K 4)
  for (int l = 0; l < 5; l++) {
    const float* hptr; int lda, Kh;
    if (ih < 0) { hptr = W + oH0; lda = 4; Kh = 4; }
    else        { hptr = W + oP[ih]; lda = FPAD; Kh = 40; }
    int ix1  = (ih < 0) ? 0 : (ih + 1) & 3;
    int ix2  = (ih < 0) ? 1 : (ih + 2) & 3;
    int ix12 = (ih < 0) ? 2 : (ih + 3) & 3;
    int ihn  = (ih < 0) ? 3 : ix1;
    k_conv<<<dim3(48, NB), 128, 0, stream>>>(hptr, lda, Kh, W + owa[l],
                                             nullptr, 0, 0, nullptr,
                                             W + oM, W + oP[ix1], 1);
    k_conv<<<dim3(48, NB), 128, 0, stream>>>(hptr, lda, Kh, W + owb[l],
                                             nullptr, 0, 0, nullptr,
                                             W + oM, W + oP[ix2], 2);
    k_bmm<<<dim3(40, NB), 128, 0, stream>>>(W + oP[ix1], W + oP[ix2], W + oM,
                                            W + oP[ix12]);
    k_conv<<<dim3(48, NB), 128, 0, stream>>>(W + oP[ix12], FPAD, 40, W + owc1[l],
                                             hptr, lda, Kh, W + owc2[l],
                                             W + oM, W + oP[ihn], 0);
    k_readout<<<NB, 128, 0, stream>>>(W + oP[ihn], Mmask, W + oOuts, l);
    ih = ihn;
  }
  // head: out[b][0..31] = relu(outs @ h1_w + h1_b)
  k_gemm_rows<2, 1, 0><<<2, 128, 0, stream>>>(W + oOuts, 400, W + oH1wT, 400,
                                              h1b, out, 96);
}